// NUFFT_19310172962928
// MI455X (gfx1250) — compile-verified
//
#include <hip/hip_runtime.h>
#include <hip/hip_bf16.h>
#include <math.h>

// ---------------- problem constants ----------------
#define NN    256
#define OSF   2
#define GG    512           // N * OSF
#define JJ    6
#define BETA  14.04f        // 2.34 * J
#define CC    12
#define KK    131072
#define G2    (GG*GG)       // 262144 complex per coil plane

typedef float v2f __attribute__((ext_vector_type(2)));
typedef float v8f __attribute__((ext_vector_type(8)));

// ---------------- device helpers ----------------

// Modified Bessel I0, Abramowitz & Stegun 9.8.1 / 9.8.2 (fp32)
__device__ __forceinline__ float bessel_i0f(float x) {
    float ax = fabsf(x);
    if (ax <= 3.75f) {
        float t = ax / 3.75f;
        float t2 = t * t;
        return 1.0f + t2*(3.5156229f + t2*(3.0899424f + t2*(1.2067492f
                   + t2*(0.2659732f + t2*(0.0360768f + t2*0.0045813f)))));
    } else {
        float it = 3.75f / ax;
        float p = 0.39894228f + it*(0.01328592f + it*(0.00225319f + it*(-0.00157565f
                + it*(0.00916281f + it*(-0.02057706f + it*(0.02635537f
                + it*(-0.01647633f + it*0.00392377f)))))));
        return (expf(ax) * rsqrtf(ax)) * p;
    }
}

// Kaiser-Bessel gridding kernel: |u|<3 -> i0(BETA*sqrt(1-(u/3)^2)), else 0
__device__ __forceinline__ float kb_kernel(float u) {
    float s = u * (2.0f / (float)JJ);        // 2u/J
    float inside = 1.0f - s * s;
    if (inside > 0.0f) return bessel_i0f(BETA * sqrtf(inside));
    return 0.0f;
}

// Fourier transform of KB kernel (apodization), t = (i - N/2)/G
__device__ __forceinline__ float kb_ft(float t) {
    float pj = (float)M_PI * (float)JJ * t;
    float w = BETA * BETA - pj * pj;
    float sw = sqrtf(fmaxf(fabsf(w), 1e-12f));
    float v;
    if (w > 0.0f) v = sinhf(sw) / sw;
    else          v = sinf(sw) / sw;
    return (float)JJ * v;
}

// ---------------- kernel 1: DFT matrix W[c,f] = exp(-2*pi*i*c*f/512) ----------------
__global__ void genW_kernel(float* __restrict__ W) {
    int t = blockIdx.x * blockDim.x + threadIdx.x;
    if (t >= G2) return;
    int f = t & (GG - 1);
    int c = t >> 9;
    int p = (c * f) & (GG - 1);               // exact integer phase reduction
    float ang = -2.0f * (float)M_PI * (float)p / (float)GG;
    float sn, cs;
    __sincosf(ang, &sn, &cs);
    W[2*t + 0] = cs;
    W[2*t + 1] = sn;
}

// ---------------- kernel 2: SENSE multiply, apod correct, roll-scatter, zero-pad ----------------
// Writes the full pre-rolled padded grid X[c, a, b] (zero outside image support).
__global__ void prep_kernel(const float* __restrict__ in_re, const float* __restrict__ in_im,
                            const float* __restrict__ sm_re, const float* __restrict__ sm_im,
                            float* __restrict__ X) {
    int t = blockIdx.x * blockDim.x + threadIdx.x;
    if (t >= CC * G2) return;
    int b = t & (GG - 1);
    int a = (t >> 9) & (GG - 1);
    int c = t >> 18;
    // inverse of roll(-(N/2)): source image coords
    int i = (a + NN / 2) & (GG - 1);
    int j = (b + NN / 2) & (GG - 1);
    float re = 0.0f, im = 0.0f;
    if (i < NN && j < NN) {
        int ij = i * NN + j;
        float xr = in_re[ij], xi = in_im[ij];
        float sr = sm_re[c * NN * NN + ij], si = sm_im[c * NN * NN + ij];
        re = xr * sr - xi * si;
        im = xr * si + xi * sr;
        float ti = ((float)i - (float)(NN / 2)) / (float)GG;
        float tj = ((float)j - (float)(NN / 2)) / (float)GG;
        float ap = kb_ft(ti) * kb_ft(tj);
        float sc = 1.0f / (ap * (float)GG);   // fold in 1/sqrt(G*G)
        re *= sc;
        im *= sc;
    }
    float* dst = X + 2 * (size_t)t;
    dst[0] = re;
    dst[1] = im;
}

// ---------------- kernel 3: complex 512^3 GEMM via V_WMMA_F32_16X16X4_F32 ----------------
// D = A * B (row-major complex interleaved). One wave computes a 32x32 macro-tile
// (2x2 register-blocked 16x16 tiles) so each A fragment feeds 2 column tiles and
// each B fragment feeds 2 row tiles (halves load bytes per WMMA vs 1 tile/wave).
// blockIdx.z selects the batch (coil); strides are in complex elements.
__global__ void zgemm512_wmma_kernel(const float* __restrict__ Abase, long strideA,
                                     const float* __restrict__ Bbase, long strideB,
                                     float* __restrict__ Dbase, long strideD) {
    const float* A = Abase + 2l * strideA * blockIdx.z;
    const float* B = Bbase + 2l * strideB * blockIdx.z;
    float*       D = Dbase + 2l * strideD * blockIdx.z;

    const int lane = threadIdx.x;       // 0..31
    const int half = lane >> 4;         // A/B K-pair selector per lane half
    const int l    = lane & 15;
    const int row0 = blockIdx.x * 32;
    const int col0 = blockIdx.y * 32;

    v8f accr[2][2] = {};                // [row tile][col tile] real accums
    v8f acci[2][2] = {};                // imag accums

    for (int k0 = 0; k0 < GG; k0 += 4) {
        const int ca = k0 + 2 * half;

        // A fragments for 2 row tiles: two consecutive complex -> float4 {re0,im0,re1,im1}
        v2f ar[2], ai[2];
#pragma unroll
        for (int rt = 0; rt < 2; ++rt) {
            int r = row0 + 16 * rt + l;
            float4 av = *(const float4*)(A + 2 * (r * GG + ca));
            ar[rt] = (v2f){ av.x, av.z };
            ai[rt] = (v2f){ av.y, av.w };
        }

        // B fragments for 2 column tiles: rows ca, ca+1 at column n
        v2f br[2], bi[2], nbi[2];
#pragma unroll
        for (int ct = 0; ct < 2; ++ct) {
            int n = col0 + 16 * ct + l;
            float2 b0 = *(const float2*)(B + 2 * (ca * GG + n));
            float2 b1 = *(const float2*)(B + 2 * ((ca + 1) * GG + n));
            br[ct]  = (v2f){  b0.x,  b1.x };
            bi[ct]  = (v2f){  b0.y,  b1.y };
            nbi[ct] = (v2f){ -b0.y, -b1.y };
        }

        // Re += Ar*Br - Ai*Bi ; Im += Ar*Bi + Ai*Br   (16 WMMAs per K-step)
#pragma unroll
        for (int rt = 0; rt < 2; ++rt) {
#pragma unroll
            for (int ct = 0; ct < 2; ++ct) {
                accr[rt][ct] = __builtin_amdgcn_wmma_f32_16x16x4_f32(
                    false, ar[rt], false, br[ct],  (short)0, accr[rt][ct], false, false);
                accr[rt][ct] = __builtin_amdgcn_wmma_f32_16x16x4_f32(
                    false, ai[rt], false, nbi[ct], (short)0, accr[rt][ct], false, false);
                acci[rt][ct] = __builtin_amdgcn_wmma_f32_16x16x4_f32(
                    false, ar[rt], false, bi[ct],  (short)0, acci[rt][ct], false, false);
                acci[rt][ct] = __builtin_amdgcn_wmma_f32_16x16x4_f32(
                    false, ai[rt], false, br[ct],  (short)0, acci[rt][ct], false, false);
            }
        }
    }

    // C/D layout: VGPR v, lanes 0-15 -> M=v, lanes 16-31 -> M=v+8 ; N = lane%16
#pragma unroll
    for (int rt = 0; rt < 2; ++rt) {
#pragma unroll
        for (int ct = 0; ct < 2; ++ct) {
#pragma unroll
            for (int v = 0; v < 8; ++v) {
                int row = row0 + 16 * rt + v + 8 * half;
                int col = col0 + 16 * ct + l;
                float2 o = { accr[rt][ct][v], acci[rt][ct][v] };
                *(float2*)(D + 2 * (row * GG + col)) = o;
            }
        }
    }
}

// ---------------- kernel 4: gridding interpolation ----------------
// One thread per trajectory point; 12 coils x 6x6 separable KB gather from L2-resident grid.
__global__ void interp_kernel(const float* __restrict__ Y,      // 12 x 512 x 512 complex
                              const float* __restrict__ ktraj,  // (2, K)
                              const float* __restrict__ dcomp,  // (1, K)
                              float* __restrict__ out) {        // (C, K) complex interleaved
    int k = blockIdx.x * blockDim.x + threadIdx.x;
    if (k >= KK) return;

    // om = roll(ktraj, 1, axis=1): row0 <- ktraj row1 (x/ix), row1 <- ktraj row0 (y/iy)
    float gx = ktraj[KK + k] * (float)GG;
    float gy = ktraj[k]      * (float)GG;
    gx -= floorf(gx / (float)GG) * (float)GG;   // % G
    gy -= floorf(gy / (float)GG) * (float)GG;

    float bxf = floorf(gx);
    float byf = floorf(gy);
    int bx = (int)bxf;
    int by = (int)byf;
    float fx = gx - bxf;
    float fy = gy - byf;

    int   ix[JJ], iy[JJ];
    float wx[JJ], wy[JJ];
#pragma unroll
    for (int t = 0; t < JJ; ++t) {
        int off = t - (JJ / 2 - 1);             // -2..3
        ix[t] = (bx + off + GG) & (GG - 1);
        iy[t] = (by + off + GG) & (GG - 1);
        wx[t] = kb_kernel(fx - (float)off);
        wy[t] = kb_kernel(fy - (float)off);
    }

    float sd = sqrtf(dcomp[k]);

    for (int c = 0; c < CC; ++c) {
        const float2* g = (const float2*)(Y + 2 * (size_t)c * G2);
        float re = 0.0f, im = 0.0f;
#pragma unroll
        for (int i = 0; i < JJ; ++i) {
            int rowoff = ix[i] * GG;
            float rre = 0.0f, rim = 0.0f;
#pragma unroll
            for (int j = 0; j < JJ; ++j) {
                float2 v = g[rowoff + iy[j]];
                rre += wy[j] * v.x;
                rim += wy[j] * v.y;
            }
            re += wx[i] * rre;
            im += wx[i] * rim;
        }
        size_t o = 2 * ((size_t)c * KK + (size_t)k);
        out[o + 0] = re * sd;
        out[o + 1] = im * sd;
    }
}

// ---------------- host launcher ----------------
extern "C" void kernel_launch(void* const* d_in, const int* in_sizes, int n_in,
                              void* d_out, int out_size, void* d_ws, size_t ws_size,
                              hipStream_t stream) {
    const float* in_re = (const float*)d_in[0];
    const float* in_im = (const float*)d_in[1];
    const float* sm_re = (const float*)d_in[2];
    const float* sm_im = (const float*)d_in[3];
    const float* ktraj = (const float*)d_in[4];
    const float* dcomp = (const float*)d_in[5];
    float* out = (float*)d_out;

    char* ws = (char*)d_ws;
    const size_t planeBytes = (size_t)G2 * 2 * sizeof(float);       // 2 MB per complex plane
    float* W  = (float*)(ws);                                       // 1 plane   (2 MB)
    float* X  = (float*)(ws + planeBytes);                          // 12 planes (24 MB)
    float* M1 = (float*)(ws + planeBytes * (1 + CC));               // 12 planes (24 MB)
    float* Y  = (float*)(ws + planeBytes * (1 + 2 * CC));           // 12 planes (24 MB)

    // 1) DFT matrix
    {
        int total = G2;
        genW_kernel<<<(total + 255) / 256, 256, 0, stream>>>(W);
    }
    // 2) SENSE + apodization + roll + zero-pad
    {
        int total = CC * G2;
        prep_kernel<<<(total + 255) / 256, 256, 0, stream>>>(in_re, in_im, sm_re, sm_im, X);
    }
    // 3) M1[c] = X[c] * W   (row transform)
    {
        dim3 grid(GG / 32, GG / 32, CC);
        zgemm512_wmma_kernel<<<grid, 32, 0, stream>>>(X, (long)G2, W, 0l, M1, (long)G2);
    }
    // 4) Y[c] = W * M1[c]   (column transform; W symmetric)
    {
        dim3 grid(GG / 32, GG / 32, CC);
        zgemm512_wmma_kernel<<<grid, 32, 0, stream>>>(W, 0l, M1, (long)G2, Y, (long)G2);
    }
    // 5) gridding interpolation -> kdata * sqrt(dcomp)
    {
        interp_kernel<<<KK / 256, 256, 0, stream>>>(Y, ktraj, dcomp, out);
    }
}